// OnlineLayer_76639396430210
// MI455X (gfx1250) — compile-verified
//
#include <hip/hip_runtime.h>
#include <stdint.h>

typedef float v2f __attribute__((ext_vector_type(2)));
typedef float v8f __attribute__((ext_vector_type(8)));
typedef int   b128vec __attribute__((vector_size(16)));   // matches builtin param

#define V      8192
#define CIN    64
#define COUT   32
#define NP     3
#define NELT   (COUT * V)          // 262144
#define WTILE  128
#define KTILE  64
#define ASTRIDE (WTILE + 4)        // 132 floats: 528B rows (16B aligned), pads banks
#define HSTRIDE (KTILE + 4)        // 68 floats: 272B rows (16B aligned)
#define KCHUNKS 6                  // p(3) x v-halves(2); K per chunk = 4096
#define EPSLN  1e-5f

__device__ __forceinline__ v8f wmma_f32(v2f a, v2f b, v8f c) {
  // (neg_a, A, neg_b, B, c_mod, C, reuse_a, reuse_b)
  return __builtin_amdgcn_wmma_f32_16x16x4_f32(false, a, false, b, (short)0, c,
                                               false, false);
}

// -------- gfx1250 async global->LDS copy (ASYNCcnt-tracked) ----------------
__device__ __forceinline__ void async_copy_b128(const float* g, float* lds) {
#if __has_builtin(__builtin_amdgcn_global_load_async_to_lds_b128)
  __builtin_amdgcn_global_load_async_to_lds_b128(
      (__attribute__((address_space(1))) b128vec*)(uintptr_t)g,
      (__attribute__((address_space(3))) b128vec*)(uint32_t)(uintptr_t)lds,
      0, 0);
#else
  const uint32_t lds_off = (uint32_t)(uintptr_t)lds;  // generic LDS addr[31:0]
  asm volatile("global_load_async_to_lds_b128 %0, %1, off"
               :: "v"(lds_off), "v"(g) : "memory");
#endif
}

__device__ __forceinline__ void wait_async_le10() {
#if __has_builtin(__builtin_amdgcn_s_wait_asynccnt)
  __builtin_amdgcn_s_wait_asynccnt(10);
#else
  asm volatile("s_wait_asynccnt 0xa" ::: "memory");
#endif
}

__device__ __forceinline__ void wait_async_0() {
#if __has_builtin(__builtin_amdgcn_s_wait_asynccnt)
  __builtin_amdgcn_s_wait_asynccnt(0);
#else
  asm volatile("s_wait_asynccnt 0x0" ::: "memory");
#endif
}

// ---------------------------------------------------------------------------
// K1: h = conv_w @ x + conv_b (96 x 8192); res_pre = res_w @ x (32 x 8192)
//     + per-block partial sum/sumsq of res_pre for its LayerNorm.
// ---------------------------------------------------------------------------
__global__ __launch_bounds__(256) void k1_conv(
    const float* __restrict__ x, const float* __restrict__ conv_w,
    const float* __restrict__ conv_b, const float* __restrict__ res_w,
    float* __restrict__ h, float* __restrict__ res_pre,
    float* __restrict__ pres)
{
  const int t = threadIdx.x;
  const int w = blockIdx.x * 256 + t;

  float xr[CIN];
#pragma unroll
  for (int c = 0; c < CIN; ++c) xr[c] = x[c * V + w];

  for (int o = 0; o < NP * COUT; ++o) {
    float acc = conv_b[o];
#pragma unroll
    for (int c = 0; c < CIN; ++c) acc = fmaf(conv_w[o * CIN + c], xr[c], acc);
    h[(size_t)o * V + w] = acc;
  }

  float s = 0.f, s2 = 0.f;
  for (int o = 0; o < COUT; ++o) {
    float acc = 0.f;
#pragma unroll
    for (int c = 0; c < CIN; ++c) acc = fmaf(res_w[o * CIN + c], xr[c], acc);
    res_pre[(size_t)o * V + w] = acc;
    s += acc; s2 += acc * acc;
  }

  __shared__ float r1[256], r2[256];
  r1[t] = s; r2[t] = s2;
  __syncthreads();
  for (int off = 128; off > 0; off >>= 1) {
    if (t < off) { r1[t] += r1[t + off]; r2[t] += r2[t + off]; }
    __syncthreads();
  }
  if (t == 0) { pres[2 * blockIdx.x] = r1[0]; pres[2 * blockIdx.x + 1] = r2[0]; }
}

// ---------------------------------------------------------------------------
// K2: y_part[chunk] = h[p] @ A[p] over this chunk's K range.
//     v_wmma_f32_16x16x4_f32; double-buffered LDS fed by async load-to-LDS.
//     Block = 256 thr (8 waves), 128 w-columns; wave w owns cols [16w,16w+16).
// ---------------------------------------------------------------------------
__global__ __launch_bounds__(256) void k2_matmul(
    const float* __restrict__ A, const float* __restrict__ h,
    float* __restrict__ ypart)
{
  __shared__ float Alds[2][KTILE * ASTRIDE];   // 2 x 33 KB
  __shared__ float Hlds[2][COUT * HSTRIDE];    // 2 x 8.5 KB

  const int t    = threadIdx.x;
  const int lane = t & 31;
  const int wave = t >> 5;
  const int kgrp = lane >> 4;     // which K pair this lane holds (0 or 1)
  const int lid  = lane & 15;
  const int wcol = (wave << 4) + lid;

  const int w0 = blockIdx.x * WTILE;
  const int p  = blockIdx.y >> 1;
  const int v0 = (blockIdx.y & 1) * 4096;

  const float* Ab = A + ((size_t)p * V + v0) * V + w0;
  const float* Hb = h + (size_t)p * COUT * V + v0;

  // this thread's 10 async-copy slots (8 for A tile, 2 for H tile)
  const int ar[8] = { (t + 0*256) >> 5, (t + 1*256) >> 5, (t + 2*256) >> 5,
                      (t + 3*256) >> 5, (t + 4*256) >> 5, (t + 5*256) >> 5,
                      (t + 6*256) >> 5, (t + 7*256) >> 5 };
  const int ac = (t & 31) * 4;                  // 32 float4 per 128-wide row
  const int hr0 = t >> 4, hr1 = (t + 256) >> 4; // 16 float4 per 64-wide row
  const int hc = (t & 15) * 4;

  v8f acc0 = {}; v8f acc1 = {};

  // issue tile 0 into buffer 0
#pragma unroll
  for (int i = 0; i < 8; ++i)
    async_copy_b128(Ab + (size_t)ar[i] * V + ac, &Alds[0][ar[i] * ASTRIDE + ac]);
  async_copy_b128(Hb + (size_t)hr0 * V + hc, &Hlds[0][hr0 * HSTRIDE + hc]);
  async_copy_b128(Hb + (size_t)hr1 * V + hc, &Hlds[0][hr1 * HSTRIDE + hc]);

  int buf = 0;
  for (int vt = 0; vt < 4096; vt += KTILE) {
    const bool has_next = (vt + KTILE) < 4096;
    if (has_next) {             // background-copy tile vt+1 into other buffer
      const int vn = vt + KTILE;
      const int nb = buf ^ 1;
#pragma unroll
      for (int i = 0; i < 8; ++i)
        async_copy_b128(Ab + (size_t)(vn + ar[i]) * V + ac,
                        &Alds[nb][ar[i] * ASTRIDE + ac]);
      async_copy_b128(Hb + (size_t)hr0 * V + vn + hc,
                      &Hlds[nb][hr0 * HSTRIDE + hc]);
      async_copy_b128(Hb + (size_t)hr1 * V + vn + hc,
                      &Hlds[nb][hr1 * HSTRIDE + hc]);
      wait_async_le10();        // in-order: first 10 (tile vt) are complete
    } else {
      wait_async_0();
    }
    __syncthreads();            // tile vt visible from all waves

    const float* __restrict__ Al = Alds[buf];
    const float* __restrict__ Hl = Hlds[buf];
#pragma unroll
    for (int kk = 0; kk < KTILE; kk += 4) {
      const int kA = kk + (kgrp << 1);
      v2f b;                    // B frag 4x16: rows kA,kA+1 at column wcol
      b[0] = Al[kA * ASTRIDE + wcol];
      b[1] = Al[(kA + 1) * ASTRIDE + wcol];
      v2f a0, a1;               // A frags 16x4: row lid / lid+16, K=kA,kA+1
      a0[0] = Hl[lid * HSTRIDE + kA];
      a0[1] = Hl[lid * HSTRIDE + kA + 1];
      a1[0] = Hl[(lid + 16) * HSTRIDE + kA];
      a1[1] = Hl[(lid + 16) * HSTRIDE + kA + 1];
      acc0 = wmma_f32(a0, b, acc0);
      acc1 = wmma_f32(a1, b, acc1);
    }
    __syncthreads();            // all waves done with buf before it is refilled
    buf ^= 1;
  }

  // D layout: VGPR r -> M=r (lanes 0-15) / M=r+8 (lanes 16-31), N=lid
  float* yp = ypart + (size_t)blockIdx.y * NELT;
  const int n = w0 + wcol;
#pragma unroll
  for (int r = 0; r < 8; ++r) {
    const int c = r + (kgrp << 3);
    yp[(size_t)c * V + n]        = acc0[r];
    yp[(size_t)(c + 16) * V + n] = acc1[r];
  }
}

// ---------------------------------------------------------------------------
// K3: y = sum of 6 partial slabs (fixed order); agg = acc + y - fifo;
//     per-block partial sum/sumsq of agg.
// ---------------------------------------------------------------------------
__global__ __launch_bounds__(256) void k3_agg(
    const float* __restrict__ ypart, const float* __restrict__ acc_slot,
    const float* __restrict__ fifo, float* __restrict__ agg,
    float* __restrict__ pagg)
{
  const int t = threadIdx.x;
  const int i = blockIdx.x * 256 + t;
  float yv = 0.f;
#pragma unroll
  for (int ch = 0; ch < KCHUNKS; ++ch) yv += ypart[(size_t)ch * NELT + i];
  const float a = acc_slot[i] + yv - fifo[i];
  agg[i] = a;

  __shared__ float r1[256], r2[256];
  r1[t] = a; r2[t] = a * a;
  __syncthreads();
  for (int off = 128; off > 0; off >>= 1) {
    if (t < off) { r1[t] += r1[t + off]; r2[t] += r2[t + off]; }
    __syncthreads();
  }
  if (t == 0) { pagg[2 * blockIdx.x] = r1[0]; pagg[2 * blockIdx.x + 1] = r2[0]; }
}

// ---------------------------------------------------------------------------
// K4: finalize LayerNorm stats (deterministic fixed-order sums).
// ---------------------------------------------------------------------------
__global__ void k4_stats(const float* __restrict__ pres,
                         const float* __restrict__ pagg,
                         float* __restrict__ st)
{
  if (threadIdx.x == 0) {
    const float N = (float)NELT;
    float s = 0.f, s2 = 0.f;
    for (int i = 0; i < NELT / 256; ++i) { s += pagg[2 * i]; s2 += pagg[2 * i + 1]; }
    float m = s / N, var = s2 / N - m * m;
    st[0] = m; st[1] = rsqrtf(var + EPSLN);
    s = 0.f; s2 = 0.f;
    for (int i = 0; i < V / 256; ++i) { s += pres[2 * i]; s2 += pres[2 * i + 1]; }
    m = s / N; var = s2 / N - m * m;
    st[2] = m; st[3] = rsqrtf(var + EPSLN);
  }
}

// ---------------------------------------------------------------------------
// K5: out = relu( relu(LN(agg)) + LN(res_pre) )
// ---------------------------------------------------------------------------
__global__ __launch_bounds__(256) void k5_out(
    const float* __restrict__ agg, const float* __restrict__ resp,
    const float* __restrict__ lnw, const float* __restrict__ lnb,
    const float* __restrict__ rlnw, const float* __restrict__ rlnb,
    const float* __restrict__ st, float* __restrict__ out)
{
  const int i = blockIdx.x * 256 + threadIdx.x;
  const float m_a = st[0], rs_a = st[1], m_r = st[2], rs_r = st[3];
  float a = (agg[i] - m_a) * rs_a * lnw[i] + lnb[i];
  a = fmaxf(a, 0.f);
  const float r = (resp[i] - m_r) * rs_r * rlnw[i] + rlnb[i];
  out[i] = fmaxf(a + r, 0.f);
}

// ---------------------------------------------------------------------------
extern "C" void kernel_launch(void* const* d_in, const int* in_sizes, int n_in,
                              void* d_out, int out_size, void* d_ws, size_t ws_size,
                              hipStream_t stream)
{
  const float* x      = (const float*)d_in[0];
  const float* A      = (const float*)d_in[1];
  const float* conv_w = (const float*)d_in[2];
  const float* conv_b = (const float*)d_in[3];
  const float* fifo   = (const float*)d_in[4];
  const float* acc    = (const float*)d_in[5];
  const float* ln_w   = (const float*)d_in[6];
  const float* ln_b   = (const float*)d_in[7];
  const float* res_w  = (const float*)d_in[8];
  const float* rln_w  = (const float*)d_in[9];
  const float* rln_b  = (const float*)d_in[10];
  float* out = (float*)d_out;

  float* ws    = (float*)d_ws;
  float* h     = ws;                              // 96*8192
  float* resp  = h + (size_t)NP * COUT * V;       // 262144
  float* ypart = resp + NELT;                     // 6*262144
  float* agg   = ypart + (size_t)KCHUNKS * NELT;  // 262144
  float* pres  = agg + NELT;                      // 64
  float* pagg  = pres + 2 * (V / 256);            // 2048
  float* st    = pagg + 2 * (NELT / 256);         // 4

  k1_conv  <<<dim3(V / 256), 256, 0, stream>>>(x, conv_w, conv_b, res_w, h, resp, pres);
  k2_matmul<<<dim3(V / WTILE, KCHUNKS), 256, 0, stream>>>(A, h, ypart);
  k3_agg   <<<dim3(NELT / 256), 256, 0, stream>>>(ypart, acc, fifo, agg, pagg);
  k4_stats <<<dim3(1), 32, 0, stream>>>(pres, pagg, st);
  k5_out   <<<dim3(NELT / 256), 256, 0, stream>>>(agg, resp, ln_w, ln_b, rln_w, rln_b, st, out);

  (void)in_sizes; (void)n_in; (void)out_size; (void)ws_size;
}